// Freia_INN_44607530336409
// MI455X (gfx1250) — compile-verified
//
#include <hip/hip_runtime.h>

typedef __attribute__((ext_vector_type(2))) float v2f;
typedef __attribute__((ext_vector_type(8))) float v8f;

#define HALFD 32
#define HDIM  90
#define ODIM  64
#define SFC   1.5f
#define NBLK  6

// weight-fragment layout sizes (in float2 units)
#define L1_F2 (6 * 8  * 32)   // 1536
#define L2_F2 (6 * 24 * 32)   // 4608
#define L3_F2 (4 * 24 * 32)   // 3072
#define MLP_F2 (L1_F2 + L2_F2 + L3_F2)   // 9216
#define WS_W_F2 (12 * MLP_F2)            // 110592 float2 = 884736 B

// ---------------------------------------------------------------------------
// Prep: swizzle weights into B-fragment order (padded to K=96/N=96) + invperms
// ---------------------------------------------------------------------------
__global__ void prep_kernel(const float* __restrict__ w1a, const float* __restrict__ w2a,
                            const float* __restrict__ w3a, const float* __restrict__ w1b,
                            const float* __restrict__ w2b, const float* __restrict__ w3b,
                            const int* __restrict__ perms,
                            float2* __restrict__ wsW, int* __restrict__ wsInv) {
  int idx = blockIdx.x * 256 + threadIdx.x;
  if (idx < WS_W_F2) {
    int s   = idx / MLP_F2;          // 0..11  (block*2 + net)
    int rem = idx % MLP_F2;
    int b   = s >> 1;
    int net = s & 1;                 // 0 = nn2 (applied first), 1 = nn1
    const float* W; int Kact, Nact, ld, nt, ks, lane;
    if (rem < L1_F2) {
      nt = rem / 256; ks = (rem % 256) / 32; lane = rem % 32;
      W = (net == 0 ? w1b : w1a) + b * HALFD * HDIM; Kact = HALFD; Nact = HDIM; ld = HDIM;
    } else if (rem < L1_F2 + L2_F2) {
      int q = rem - L1_F2;
      nt = q / 768; ks = (q % 768) / 32; lane = q % 32;
      W = (net == 0 ? w2b : w2a) + b * HDIM * HDIM; Kact = HDIM; Nact = HDIM; ld = HDIM;
    } else {
      int q = rem - (L1_F2 + L2_F2);
      nt = q / 768; ks = (q % 768) / 32; lane = q % 32;
      W = (net == 0 ? w3b : w3a) + b * HDIM * ODIM; Kact = HDIM; Nact = ODIM; ld = ODIM;
    }
    int hi = lane >> 4;
    int n  = nt * 16 + (lane & 15);
    int k0 = 4 * ks + 2 * hi;
    float x0 = (k0     < Kact && n < Nact) ? W[(size_t)k0       * ld + n] : 0.f;
    float x1 = (k0 + 1 < Kact && n < Nact) ? W[(size_t)(k0 + 1) * ld + n] : 0.f;
    wsW[idx] = make_float2(x0, x1);
  } else if (idx < WS_W_F2 + NBLK * ODIM) {
    int t = idx - WS_W_F2;
    int b = t / ODIM, i = t % ODIM;
    wsInv[b * ODIM + perms[b * ODIM + i]] = i;
  }
}

// ---------------------------------------------------------------------------
// Main kernel helpers
// ---------------------------------------------------------------------------
__device__ __forceinline__ v8f wmma_f32(v2f a, v2f b, v8f c) {
  return __builtin_amdgcn_wmma_f32_16x16x4_f32(false, a, false, b, (short)0, c, false, false);
}

__device__ __forceinline__ void lds_fence() {
  asm volatile("s_wait_dscnt 0x0" ::: "memory");
}

// one 16x16 output tile: C += A(16xK) * B(Kx16), A from LDS (row-major), B from
// pre-swizzled global fragments (one b64 per lane per k-step)
__device__ __forceinline__ v8f layer_tile(const float* inlds, int instride, int ksteps,
                                          const v2f* __restrict__ wfrag, float bv, int lane) {
  int hi = lane >> 4, nl = lane & 15;
  v8f acc;
  #pragma unroll
  for (int r = 0; r < 8; ++r) acc[r] = bv;
  const float* ain = inlds + nl * instride + 2 * hi;
  for (int ks = 0; ks < ksteps; ++ks) {
    v2f a = *(const v2f*)(ain + 4 * ks);          // ds_load_b64 (8B aligned)
    v2f b = wfrag[ks * 32 + lane];                // global_load_b64, coalesced 256B
    acc = wmma_f32(a, b, acc);
  }
  return acc;
}

// 32 -> 96 -> 96 -> 64 MLP on one wave's 16-row tile
__device__ __forceinline__ void mlp_fwd(const v2f* __restrict__ frag,
                                        const float* __restrict__ bias1,
                                        const float* __restrict__ bias2,
                                        const float* __restrict__ bias3,
                                        const float* inbuf, float* h1, float* h2,
                                        v8f o[4], int lane) {
  int hi = lane >> 4, nl = lane & 15;
  // layer 1: K=32, N=96(pad)
  for (int nt = 0; nt < 6; ++nt) {
    int n = nt * 16 + nl;
    float bv = (n < HDIM) ? bias1[n] : 0.f;
    v8f acc = layer_tile(inbuf, 36, 8, frag + nt * 256, bv, lane);
    #pragma unroll
    for (int r = 0; r < 8; ++r) { float v = acc[r]; acc[r] = v > 0.f ? v : 0.01f * v; }
    #pragma unroll
    for (int r = 0; r < 8; ++r) h1[(r + 8 * hi) * 100 + n] = acc[r];
  }
  lds_fence();
  // layer 2: K=96, N=96
  const v2f* f2 = frag + L1_F2;
  for (int nt = 0; nt < 6; ++nt) {
    int n = nt * 16 + nl;
    float bv = (n < HDIM) ? bias2[n] : 0.f;
    v8f acc = layer_tile(h1, 100, 24, f2 + nt * 768, bv, lane);
    #pragma unroll
    for (int r = 0; r < 8; ++r) { float v = acc[r]; acc[r] = v > 0.f ? v : 0.01f * v; }
    #pragma unroll
    for (int r = 0; r < 8; ++r) h2[(r + 8 * hi) * 100 + n] = acc[r];
  }
  lds_fence();
  // layer 3: K=96, N=64 (result stays in fragments)
  const v2f* f3 = frag + L1_F2 + L2_F2;
  for (int nt = 0; nt < 4; ++nt) {
    float bv = bias3[nt * 16 + nl];
    o[nt] = layer_tile(h2, 100, 24, f3 + nt * 768, bv, lane);
  }
  lds_fence();
}

// ---------------------------------------------------------------------------
// Main kernel: 4 waves / block, each wave owns 16 batch rows end-to-end
// ---------------------------------------------------------------------------
__global__ __launch_bounds__(128)
void inn_kernel(const float* __restrict__ x,
                const float* __restrict__ b1a, const float* __restrict__ b2a,
                const float* __restrict__ b3a, const float* __restrict__ b1b,
                const float* __restrict__ b2b, const float* __restrict__ b3b,
                const v2f* __restrict__ wsW, const int* __restrict__ wsInv,
                float* __restrict__ out, int batch) {
  __shared__ float hA[4][16][100];
  __shared__ float hB[4][16][100];
  __shared__ float ab[4][16][36];

  const int lane = threadIdx.x & 31;
  const int wv   = threadIdx.x >> 5;
  const int hi   = lane >> 4;
  const int nl   = lane & 15;
  const int row0 = (blockIdx.x * 4 + wv) * 16;

  float* hAp = &hA[wv][0][0];
  float* hBp = &hB[wv][0][0];
  float* abp = &ab[wv][0][0];

  // load z fragments (C-layout: m = r + 8*hi, n = nl + 16*t)
  v8f zc[4], ldacc;
  #pragma unroll
  for (int r = 0; r < 8; ++r) ldacc[r] = 0.f;
  #pragma unroll
  for (int t = 0; t < 4; ++t)
    #pragma unroll
    for (int r = 0; r < 8; ++r)
      zc[t][r] = x[(size_t)(row0 + r + 8 * hi) * ODIM + t * 16 + nl];

  for (int blk = 0; blk < NBLK; ++blk) {
    const v2f* fragB = wsW + (size_t)(blk * 2 + 0) * MLP_F2;  // nn2
    const v2f* fragA = wsW + (size_t)(blk * 2 + 1) * MLP_F2;  // nn1
    v8f o[4], o2[4], y1c[2], y2c[2];

    // ---- MLP on x2 (nn2) ----
    #pragma unroll
    for (int t = 0; t < 2; ++t)
      #pragma unroll
      for (int r = 0; r < 8; ++r)
        abp[(r + 8 * hi) * 36 + t * 16 + nl] = zc[2 + t][r];
    lds_fence();
    mlp_fwd(fragB, b1b + blk * HDIM, b2b + blk * HDIM, b3b + blk * ODIM,
            abp, hAp, hBp, o, lane);

    #pragma unroll
    for (int t = 0; t < 4; ++t)
      #pragma unroll
      for (int r = 0; r < 8; ++r) o[t][r] = tanhf(o[t][r]) * SFC;   // ls2 | t2

    #pragma unroll
    for (int j = 0; j < 2; ++j)
      #pragma unroll
      for (int r = 0; r < 8; ++r) {
        y1c[j][r] = zc[j][r] * expf(o[j][r]) + o[j + 2][r];
        ldacc[r] += o[j][r];
      }

    // ---- MLP on y1 (nn1) ----
    #pragma unroll
    for (int t = 0; t < 2; ++t)
      #pragma unroll
      for (int r = 0; r < 8; ++r)
        abp[(r + 8 * hi) * 36 + t * 16 + nl] = y1c[t][r];
    lds_fence();
    mlp_fwd(fragA, b1a + blk * HDIM, b2a + blk * HDIM, b3a + blk * ODIM,
            abp, hAp, hBp, o2, lane);

    #pragma unroll
    for (int t = 0; t < 4; ++t)
      #pragma unroll
      for (int r = 0; r < 8; ++r) o2[t][r] = tanhf(o2[t][r]) * SFC; // ls1 | t1

    #pragma unroll
    for (int j = 0; j < 2; ++j)
      #pragma unroll
      for (int r = 0; r < 8; ++r) {
        y2c[j][r] = zc[2 + j][r] * expf(o2[j][r]) + o2[j + 2][r];
        ldacc[r] += o2[j][r];
      }

    // ---- permutation: scatter through LDS with inverse perm, re-gather ----
    const int* invp = wsInv + blk * ODIM;
    int cn[4];
    #pragma unroll
    for (int t = 0; t < 4; ++t) cn[t] = invp[t * 16 + nl];
    #pragma unroll
    for (int t = 0; t < 2; ++t)
      #pragma unroll
      for (int r = 0; r < 8; ++r)
        hAp[(r + 8 * hi) * 100 + cn[t]] = y1c[t][r];
    #pragma unroll
    for (int t = 2; t < 4; ++t)
      #pragma unroll
      for (int r = 0; r < 8; ++r)
        hAp[(r + 8 * hi) * 100 + cn[t]] = y2c[t - 2][r];
    lds_fence();
    #pragma unroll
    for (int t = 0; t < 4; ++t)
      #pragma unroll
      for (int r = 0; r < 8; ++r)
        zc[t][r] = hAp[(r + 8 * hi) * 100 + t * 16 + nl];
    lds_fence();
  }

  // ---- store z ----
  #pragma unroll
  for (int t = 0; t < 4; ++t)
    #pragma unroll
    for (int r = 0; r < 8; ++r)
      out[(size_t)(row0 + r + 8 * hi) * ODIM + t * 16 + nl] = zc[t][r];

  // ---- reduce + store log_det (rows m = r + 8*hi) ----
  float* ldo = out + (size_t)batch * ODIM;
  #pragma unroll
  for (int r = 0; r < 8; ++r) {
    float v = ldacc[r];
    v += __shfl_xor(v, 1);
    v += __shfl_xor(v, 2);
    v += __shfl_xor(v, 4);
    v += __shfl_xor(v, 8);
    if (nl == 0) ldo[row0 + r + 8 * hi] = v;
  }
}

// ---------------------------------------------------------------------------
extern "C" void kernel_launch(void* const* d_in, const int* in_sizes, int n_in,
                              void* d_out, int out_size, void* d_ws, size_t ws_size,
                              hipStream_t stream) {
  const float* x   = (const float*)d_in[0];
  const float* w1a = (const float*)d_in[1];
  const float* b1a = (const float*)d_in[2];
  const float* w2a = (const float*)d_in[3];
  const float* b2a = (const float*)d_in[4];
  const float* w3a = (const float*)d_in[5];
  const float* b3a = (const float*)d_in[6];
  const float* w1b = (const float*)d_in[7];
  const float* b1b = (const float*)d_in[8];
  const float* w2b = (const float*)d_in[9];
  const float* b2b = (const float*)d_in[10];
  const float* w3b = (const float*)d_in[11];
  const float* b3b = (const float*)d_in[12];
  const int* perms = (const int*)d_in[13];

  float*  out   = (float*)d_out;
  float2* wsW   = (float2*)d_ws;
  int*    wsInv = (int*)((char*)d_ws + (size_t)WS_W_F2 * sizeof(float2));

  int batch = in_sizes[0] / ODIM;

  int prepN = WS_W_F2 + NBLK * ODIM;
  prep_kernel<<<(prepN + 255) / 256, 256, 0, stream>>>(w1a, w2a, w3a, w1b, w2b, w3b,
                                                       perms, wsW, wsInv);

  int tiles = batch / 16;                 // 65536 for BATCH=1M
  inn_kernel<<<tiles / 4, 128, 0, stream>>>(x, b1a, b2a, b3a, b1b, b2b, b3b,
                                            (const v2f*)wsW, wsInv, out, batch);
}